// _GAT_6133213299126
// MI455X (gfx1250) — compile-verified
//
#include <hip/hip_runtime.h>
#include <math.h>

// ---------------------------------------------------------------------------
// GAT layer for MI455X (gfx1250, wave32).
//   inputs : feats[N,64] f32, src[E] i32, dst[E] i32, W[64,64] f32,
//            attn_l[64] f32, attn_r[64] f32, bias[64] f32
//   output : rst[N,64] f32
// Projection GEMM uses V_WMMA_F32_16X16X4_F32 (fp32 matrix core).
// ---------------------------------------------------------------------------

#define N_NODES 50000
#define N_EDGES 800000
#define DIM 64

typedef __attribute__((ext_vector_type(2))) float v2f;
typedef __attribute__((ext_vector_type(8))) float v8f;

__device__ __forceinline__ float leaky(float x) {
    return x >= 0.0f ? x : 0.2f * x;
}

// ---------------------------------------------------------------------------
// Kernel 1: initialize out = bias (broadcast), emax = -inf, denom = 0
// ---------------------------------------------------------------------------
__global__ void gat_init_kernel(float* __restrict__ out,
                                const float* __restrict__ bias,
                                float* __restrict__ emax,
                                float* __restrict__ denom) {
    int idx = blockIdx.x * blockDim.x + threadIdx.x;
    if (idx < N_NODES * DIM) {
        out[idx] = bias[idx & (DIM - 1)];
    }
    if (idx < N_NODES) {
        emax[idx]  = -INFINITY;
        denom[idx] = 0.0f;
    }
}

// ---------------------------------------------------------------------------
// Kernel 2: z = leaky_relu(feats) @ W  via fp32 WMMA; fused el = z*attn_l,
// er = z*attn_r row reductions.
// Block = 128 threads (4 waves). Each wave owns one 16-row tile of z.
// ---------------------------------------------------------------------------
#define WSTR 66   // padded LDS stride for W rows (bank-conflict free)
#define HSTR 68   // padded LDS stride for h-tile rows

__global__ __launch_bounds__(128)
void gat_proj_wmma_kernel(const float* __restrict__ feats,
                          const float* __restrict__ W,
                          const float* __restrict__ attn_l,
                          const float* __restrict__ attn_r,
                          float* __restrict__ z,
                          float* __restrict__ el,
                          float* __restrict__ er) {
    __shared__ float lw[DIM * WSTR];          // W, padded rows
    __shared__ float lh[4 * 16 * HSTR];       // 4 wave-tiles of leaky(feats)

    const int tid     = threadIdx.x;
    const int wave    = tid >> 5;             // 0..3
    const int lane    = tid & 31;
    const int lane_lo = lane & 15;
    const int half    = lane >> 4;            // 0: lanes 0-15, 1: lanes 16-31

    const int n_tiles = N_NODES / 16;         // 3125 exactly
    const int tile    = blockIdx.x * 4 + wave;
    const bool active = (tile < n_tiles);
    const int row0    = tile * 16;

    // Stage W into LDS (all 128 threads, coalesced)
    for (int i = tid; i < DIM * DIM; i += 128) {
        int r = i >> 6, c = i & 63;
        lw[r * WSTR + c] = W[i];
    }
    // Stage this wave's 16x64 activation tile with leaky_relu applied
    if (active) {
        float* myh = &lh[wave * 16 * HSTR];
        for (int i = lane; i < 16 * DIM; i += 32) {
            int r = i >> 6, c = i & 63;
            myh[r * HSTR + c] = leaky(feats[(row0 + r) * DIM + c]);
        }
    }
    __syncthreads();

    if (!active) return;                      // wave-uniform: EXEC stays all-1s

    const float* myh = &lh[wave * 16 * HSTR];

    v8f acc[4];
    acc[0] = {}; acc[1] = {}; acc[2] = {}; acc[3] = {};

    // D = A(16x4) x B(4x16) + C over 16 K-steps, 4 N-tiles.
    // fp32 A layout: VGPR v, lane l -> A[M = l%16][K = v + 2*(l/16)]
    // fp32 B layout: VGPR v, lane l -> B[K = v + 2*(l/16)][N = l%16]
#pragma unroll
    for (int kb = 0; kb < 16; ++kb) {
        const int k0 = kb * 4 + 2 * half;
        v2f a;
        a.x = myh[lane_lo * HSTR + k0];
        a.y = myh[lane_lo * HSTR + k0 + 1];
#pragma unroll
        for (int nb = 0; nb < 4; ++nb) {
            v2f b;
            b.x = lw[(k0)     * WSTR + nb * 16 + lane_lo];
            b.y = lw[(k0 + 1) * WSTR + nb * 16 + lane_lo];
            acc[nb] = __builtin_amdgcn_wmma_f32_16x16x4_f32(
                false, a, false, b, (short)0, acc[nb], false, false);
        }
    }

    // C/D layout: VGPR r, lane l -> D[M = r + 8*(l/16)][N = l%16 (+16*nb)]
    // Store z tile.
#pragma unroll
    for (int nb = 0; nb < 4; ++nb) {
#pragma unroll
        for (int r = 0; r < 8; ++r) {
            z[(row0 + 8 * half + r) * DIM + nb * 16 + lane_lo] = acc[nb][r];
        }
    }

    // Fused attention projections: el[m] = sum_c z[m][c]*attn_l[c]
    float al[4], ar[4];
#pragma unroll
    for (int nb = 0; nb < 4; ++nb) {
        al[nb] = attn_l[nb * 16 + lane_lo];
        ar[nb] = attn_r[nb * 16 + lane_lo];
    }
    float pl[8], pr[8];
#pragma unroll
    for (int r = 0; r < 8; ++r) {
        pl[r] = acc[0][r] * al[0] + acc[1][r] * al[1] +
                acc[2][r] * al[2] + acc[3][r] * al[3];
        pr[r] = acc[0][r] * ar[0] + acc[1][r] * ar[1] +
                acc[2][r] * ar[2] + acc[3][r] * ar[3];
    }
    // Reduce across the 16 lanes of each half (masks < 16 stay in-half).
#pragma unroll
    for (int off = 1; off < 16; off <<= 1) {
#pragma unroll
        for (int r = 0; r < 8; ++r) {
            pl[r] += __shfl_xor(pl[r], off, 32);
            pr[r] += __shfl_xor(pr[r], off, 32);
        }
    }
    if (lane_lo == 0) {
        const int rbase = row0 + 8 * half;
#pragma unroll
        for (int r = 0; r < 8; ++r) {
            el[rbase + r] = pl[r];
            er[rbase + r] = pr[r];
        }
    }
}

// ---------------------------------------------------------------------------
// Kernel 3: per-edge logit + segment max over dst
// ---------------------------------------------------------------------------
__global__ void gat_edge_logit_kernel(const int* __restrict__ src,
                                      const int* __restrict__ dst,
                                      const float* __restrict__ el,
                                      const float* __restrict__ er,
                                      float* __restrict__ ebuf,
                                      float* __restrict__ emax) {
    int eid = blockIdx.x * blockDim.x + threadIdx.x;
    if (eid >= N_EDGES) return;
    float v = leaky(el[src[eid]] + er[dst[eid]]);
    ebuf[eid] = v;
    atomicMax(&emax[dst[eid]], v);
}

// ---------------------------------------------------------------------------
// Kernel 4: ex = exp(e - emax[dst]); denom[dst] += ex
// ---------------------------------------------------------------------------
__global__ void gat_edge_exp_kernel(const int* __restrict__ dst,
                                    float* __restrict__ ebuf,
                                    const float* __restrict__ emax,
                                    float* __restrict__ denom) {
    int eid = blockIdx.x * blockDim.x + threadIdx.x;
    if (eid >= N_EDGES) return;
    int d = dst[eid];
    float ex = expf(ebuf[eid] - emax[d]);
    ebuf[eid] = ex;
    atomicAdd(&denom[d], ex);
}

// ---------------------------------------------------------------------------
// Kernel 5: out[dst] += (ex/denom[dst]) * z[src] ; 64 threads per edge
// ---------------------------------------------------------------------------
__global__ __launch_bounds__(256)
void gat_aggregate_kernel(const int* __restrict__ src,
                          const int* __restrict__ dst,
                          const float* __restrict__ ebuf,
                          const float* __restrict__ denom,
                          const float* __restrict__ z,
                          float* __restrict__ out) {
    int eid = blockIdx.x * 4 + (threadIdx.x >> 6);
    int j   = threadIdx.x & 63;
    if (eid >= N_EDGES) return;
    int s = src[eid];
    int d = dst[eid];
    float alpha = ebuf[eid] / denom[d];        // broadcast loads (L2/L0 hit)
    atomicAdd(&out[d * DIM + j], alpha * z[s * DIM + j]);
}

// ---------------------------------------------------------------------------
extern "C" void kernel_launch(void* const* d_in, const int* in_sizes, int n_in,
                              void* d_out, int out_size, void* d_ws, size_t ws_size,
                              hipStream_t stream) {
    const float* feats  = (const float*)d_in[0];
    const int*   src    = (const int*)  d_in[1];
    const int*   dst    = (const int*)  d_in[2];
    const float* W      = (const float*)d_in[3];
    const float* attn_l = (const float*)d_in[4];
    const float* attn_r = (const float*)d_in[5];
    const float* bias   = (const float*)d_in[6];
    float* out = (float*)d_out;

    // Workspace layout (floats): z | el | er | emax | denom | ebuf
    float* ws    = (float*)d_ws;
    float* z     = ws;
    float* el    = z    + (size_t)N_NODES * DIM;
    float* er    = el   + N_NODES;
    float* emax  = er   + N_NODES;
    float* denom = emax + N_NODES;
    float* ebuf  = denom + N_NODES;

    // 1) init out/emax/denom
    {
        int total = N_NODES * DIM;
        gat_init_kernel<<<(total + 255) / 256, 256, 0, stream>>>(out, bias, emax, denom);
    }
    // 2) projection (WMMA fp32) + el/er
    {
        int n_tiles = N_NODES / 16;            // 3125
        int blocks  = (n_tiles + 3) / 4;       // 4 waves per block
        gat_proj_wmma_kernel<<<blocks, 128, 0, stream>>>(feats, W, attn_l, attn_r,
                                                         z, el, er);
    }
    // 3) edge logits + segment max
    gat_edge_logit_kernel<<<(N_EDGES + 255) / 256, 256, 0, stream>>>(src, dst, el, er,
                                                                     ebuf, emax);
    // 4) exp + segment sum
    gat_edge_exp_kernel<<<(N_EDGES + 255) / 256, 256, 0, stream>>>(dst, ebuf, emax, denom);
    // 5) weighted scatter aggregation
    gat_aggregate_kernel<<<(N_EDGES + 3) / 4, 256, 0, stream>>>(src, dst, ebuf, denom,
                                                                z, out);
}